// GCN_11579231830735
// MI455X (gfx1250) — compile-verified
//
#include <hip/hip_runtime.h>
#include <cmath>

// ---------------------------------------------------------------------------
// GCN pipeline for MI455X (gfx1250, wave32, WMMA).
// All dense GEMMs use V_WMMA_F32_16X16X4_F32 (exact fp32).
// Edge aggregations are done on raw features (GEMMs hoisted out by linearity).
// ---------------------------------------------------------------------------

typedef float v2f __attribute__((ext_vector_type(2)));
typedef float v8f __attribute__((ext_vector_type(8)));

#define LN_EPS 1e-5f
#define HIDD 100

// 16-lane (half-wave) reductions: xor masks 1,2,4,8 stay within each half.
__device__ __forceinline__ float half16_sum(float v) {
    v += __shfl_xor(v, 1, 32);
    v += __shfl_xor(v, 2, 32);
    v += __shfl_xor(v, 4, 32);
    v += __shfl_xor(v, 8, 32);
    return v;
}
__device__ __forceinline__ float half16_max(float v) {
    v = fmaxf(v, __shfl_xor(v, 1, 32));
    v = fmaxf(v, __shfl_xor(v, 2, 32));
    v = fmaxf(v, __shfl_xor(v, 4, 32));
    v = fmaxf(v, __shfl_xor(v, 8, 32));
    return v;
}

// ---------------------------------------------------------------------------
// Wave-tile GEMM: each wave computes a 16-row x (NT*16)-col tile of
//   out = epilogue(A0 @ W0 [+ A1 @ W1] + bias)
// A: [M, KD] row-major.  W: [KD, NOUT] row-major.  out: [M, NOUT].
// WMMA f32 16x16x4 operand layouts (wave32):
//   A  (16x4): lane<16 -> row=lane, k={kb,kb+1};  lane>=16 -> row=lane-16, k={kb+2,kb+3}
//   B  (4x16): lane<16 -> col=lane, k rows {kb,kb+1}; lane>=16 -> col=lane-16, k={kb+2,kb+3}
//   C/D(16x16): lanes 0-15 hold rows 0-7 (vgpr j = row j), lanes 16-31 rows 8-15
// ---------------------------------------------------------------------------
template <int KD, int NT, int NOUT, bool DUAL, bool DO_LN, bool DO_RELU, bool DO_LSM>
__global__ __launch_bounds__(256) void gemm_epilogue_kernel(
    const float* __restrict__ A0, const float* __restrict__ W0,
    const float* __restrict__ A1, const float* __restrict__ W1,
    const float* __restrict__ bias,
    const float* __restrict__ gamma, const float* __restrict__ beta,
    float* __restrict__ out, int numRowTiles)
{
    const int tileM = blockIdx.x * blockDim.y + threadIdx.y;   // wave-uniform
    if (tileM >= numRowTiles) return;                          // whole-wave exit

    const int lane = threadIdx.x;       // 0..31
    const int half = lane >> 4;         // 0: rows 0-7 / k 0-1,  1: rows 8-15 / k 2-3
    const int l15  = lane & 15;
    const int koff = half * 2;

    v8f acc[NT] = {};

    const size_t rowA = (size_t)tileM * 16 + l15;

    // --- main GEMM: A0 @ W0 ---
    {
        const float* __restrict__ aBase = A0 + rowA * KD;
        for (int kb = 0; kb < KD; kb += 4) {
            const v2f a = *reinterpret_cast<const v2f*>(aBase + kb + koff);
#pragma unroll
            for (int t = 0; t < NT; ++t) {
                const int col  = t * 16 + l15;
                const int ccol = (col < NOUT) ? col : (NOUT - 1);
                const float w0 = W0[(size_t)(kb + koff) * NOUT + ccol];
                const float w1 = W0[(size_t)(kb + koff + 1) * NOUT + ccol];
                v2f b;
                b.x = (col < NOUT) ? w0 : 0.0f;   // zero out padded columns
                b.y = (col < NOUT) ? w1 : 0.0f;
                acc[t] = __builtin_amdgcn_wmma_f32_16x16x4_f32(
                    false, a, false, b, (short)0, acc[t], false, false);
            }
        }
    }

    // --- optional second GEMM accumulated into same tile: + A1 @ W1 ---
    if constexpr (DUAL) {
        const float* __restrict__ aBase = A1 + rowA * KD;
        for (int kb = 0; kb < KD; kb += 4) {
            const v2f a = *reinterpret_cast<const v2f*>(aBase + kb + koff);
#pragma unroll
            for (int t = 0; t < NT; ++t) {
                const int col  = t * 16 + l15;
                const int ccol = (col < NOUT) ? col : (NOUT - 1);
                const float w0 = W1[(size_t)(kb + koff) * NOUT + ccol];
                const float w1 = W1[(size_t)(kb + koff + 1) * NOUT + ccol];
                v2f b;
                b.x = (col < NOUT) ? w0 : 0.0f;
                b.y = (col < NOUT) ? w1 : 0.0f;
                acc[t] = __builtin_amdgcn_wmma_f32_16x16x4_f32(
                    false, a, false, b, (short)0, acc[t], false, false);
            }
        }
    }

    // --- bias ---
#pragma unroll
    for (int t = 0; t < NT; ++t) {
        const int col  = t * 16 + l15;
        const float bv = (col < NOUT) ? bias[col] : 0.0f;
#pragma unroll
        for (int j = 0; j < 8; ++j) acc[t][j] += bv;
    }

    // --- LayerNorm (per row of NOUT valid cols) ---
    if constexpr (DO_LN) {
        float s[8], s2[8];
#pragma unroll
        for (int j = 0; j < 8; ++j) { s[j] = 0.0f; s2[j] = 0.0f; }
#pragma unroll
        for (int t = 0; t < NT; ++t) {
            const int col = t * 16 + l15;
            const bool valid = (col < NOUT);
#pragma unroll
            for (int j = 0; j < 8; ++j) {
                const float v = valid ? acc[t][j] : 0.0f;
                s[j]  += v;
                s2[j] += v * v;
            }
        }
        float mu[8], inv[8];
#pragma unroll
        for (int j = 0; j < 8; ++j) {
            const float rs  = half16_sum(s[j]);
            const float rs2 = half16_sum(s2[j]);
            const float m   = rs * (1.0f / (float)NOUT);
            const float var = rs2 * (1.0f / (float)NOUT) - m * m;
            mu[j]  = m;
            inv[j] = rsqrtf(var + LN_EPS);
        }
#pragma unroll
        for (int t = 0; t < NT; ++t) {
            const int col  = t * 16 + l15;
            const int ccol = (col < NOUT) ? col : (NOUT - 1);
            const float g  = gamma[ccol];
            const float be = beta[ccol];
#pragma unroll
            for (int j = 0; j < 8; ++j)
                acc[t][j] = (acc[t][j] - mu[j]) * inv[j] * g + be;
        }
    }

    // --- ReLU ---
    if constexpr (DO_RELU) {
#pragma unroll
        for (int t = 0; t < NT; ++t)
#pragma unroll
            for (int j = 0; j < 8; ++j) acc[t][j] = fmaxf(acc[t][j], 0.0f);
    }

    // --- log_softmax over the NOUT (=NT*16) columns of each row ---
    if constexpr (DO_LSM) {
        float mx[8], se[8];
#pragma unroll
        for (int j = 0; j < 8; ++j) mx[j] = -3.402823466e38f;
#pragma unroll
        for (int t = 0; t < NT; ++t)
#pragma unroll
            for (int j = 0; j < 8; ++j) mx[j] = fmaxf(mx[j], acc[t][j]);
#pragma unroll
        for (int j = 0; j < 8; ++j) { mx[j] = half16_max(mx[j]); se[j] = 0.0f; }
#pragma unroll
        for (int t = 0; t < NT; ++t)
#pragma unroll
            for (int j = 0; j < 8; ++j) se[j] += expf(acc[t][j] - mx[j]);
#pragma unroll
        for (int j = 0; j < 8; ++j) se[j] = logf(half16_sum(se[j]));
#pragma unroll
        for (int t = 0; t < NT; ++t)
#pragma unroll
            for (int j = 0; j < 8; ++j) acc[t][j] = acc[t][j] - mx[j] - se[j];
    }

    // --- store (rows 0-7 from lanes 0-15, rows 8-15 from lanes 16-31) ---
#pragma unroll
    for (int t = 0; t < NT; ++t) {
        const int col = t * 16 + l15;
        if (col < NOUT) {
#pragma unroll
            for (int j = 0; j < 8; ++j) {
                const int row = half * 8 + j;
                out[((size_t)tileM * 16 + row) * NOUT + col] = acc[t][j];
            }
        }
    }
}

// ---------------------------------------------------------------------------
// Elementwise / edge kernels
// ---------------------------------------------------------------------------
__global__ void fill_kernel(float* __restrict__ p, float v, int n) {
    const int i = blockIdx.x * blockDim.x + threadIdx.x;
    if (i < n) p[i] = v;
}

__global__ void deg_kernel(const int* __restrict__ tgt, float* __restrict__ deg, int E) {
    const int e = blockIdx.x * blockDim.x + threadIdx.x;
    if (e < E) atomicAdd(&deg[tgt[e]], 1.0f);
}

__global__ void rsqrt_kernel(float* __restrict__ d, int n) {
    const int i = blockIdx.x * blockDim.x + threadIdx.x;
    if (i < n) d[i] = rsqrtf(d[i]);   // deg >= 1 always (self-loop)
}

// y[i,:] = dinv[i]^2 * x[i,:]   (GCN self-loop term, pre-GEMM by linearity)
__global__ void scale_self_kernel(const float* __restrict__ x, const float* __restrict__ dinv,
                                  float* __restrict__ y, int n) {
    const int i = blockIdx.x * blockDim.x + threadIdx.x;
    if (i < n) {
        const float dv = dinv[i / HIDD];
        y[i] = dv * dv * x[i];
    }
}

// out[tgt[e],:] += x[src[e],:]            (RGCN aggregation, one wave per edge)
__global__ void scatter_add_kernel(const float* __restrict__ x, const int* __restrict__ src,
                                   const int* __restrict__ tgt, float* __restrict__ out, int E) {
    const int gid  = blockIdx.x * blockDim.x + threadIdx.x;
    const int e    = gid >> 5;
    const int lane = gid & 31;
    if (e >= E) return;
    const size_t s = (size_t)src[e] * HIDD;
    const size_t t = (size_t)tgt[e] * HIDD;
    for (int d = lane; d < HIDD; d += 32) atomicAdd(&out[t + d], x[s + d]);
}

// out[tgt[e],:] += dinv[src]*dinv[tgt] * x[src[e],:]   (GCN aggregation)
__global__ void scatter_norm_kernel(const float* __restrict__ x, const int* __restrict__ src,
                                    const int* __restrict__ tgt, const float* __restrict__ dinv,
                                    float* __restrict__ out, int E) {
    const int gid  = blockIdx.x * blockDim.x + threadIdx.x;
    const int e    = gid >> 5;
    const int lane = gid & 31;
    if (e >= E) return;
    const int si = src[e], ti = tgt[e];
    const float nrm = dinv[si] * dinv[ti];
    const size_t s = (size_t)si * HIDD;
    const size_t t = (size_t)ti * HIDD;
    for (int d = lane; d < HIDD; d += 32) atomicAdd(&out[t + d], x[s + d] * nrm);
}

// ---------------------------------------------------------------------------
// Launch
// ---------------------------------------------------------------------------
extern "C" void kernel_launch(void* const* d_in, const int* in_sizes, int n_in,
                              void* d_out, int out_size, void* d_ws, size_t ws_size,
                              hipStream_t stream) {
    const int DIN  = 500;
    const int DOUT = 64;
    const int N = in_sizes[0] / DIN;
    const int E = in_sizes[1] / 2;

    const float* X     = (const float*)d_in[0];
    const int*   src   = (const int*)d_in[1];
    const int*   tgt   = src + E;
    const float* W1    = (const float*)d_in[3];
    const float* b1    = (const float*)d_in[4];
    const float* g1    = (const float*)d_in[5];
    const float* be1   = (const float*)d_in[6];
    const float* W2    = (const float*)d_in[7];
    const float* b2    = (const float*)d_in[8];
    const float* g2    = (const float*)d_in[9];
    const float* be2   = (const float*)d_in[10];
    const float* Wrel  = (const float*)d_in[11];   // [1,100,100] -> [100,100]
    const float* Wself = (const float*)d_in[12];
    const float* rb    = (const float*)d_in[13];
    const float* c1W   = (const float*)d_in[14];
    const float* c1b   = (const float*)d_in[15];
    const float* c2W   = (const float*)d_in[16];
    const float* c2b   = (const float*)d_in[17];

    float* ws   = (float*)d_ws;
    const size_t NB = (size_t)N * HIDD;
    float* bufA = ws;            // [N,100]
    float* bufB = ws + NB;       // [N,100]
    float* bufC = ws + 2 * NB;   // [N,100]
    float* dinv = ws + 3 * NB;   // [N]

    const int rowTiles = (N + 15) / 16;
    const dim3 gBlk(32, 8);
    const int  gGrid = (rowTiles + 7) / 8;
    const int  TPB = 256;
    const int  nElem = (int)NB;

    // stage 1: bufA = relu(LN(X @ W1 + b1))
    gemm_epilogue_kernel<500, 7, 100, false, true, true, false>
        <<<gGrid, gBlk, 0, stream>>>(X, W1, nullptr, nullptr, b1, g1, be1, bufA, rowTiles);

    // stage 2: bufB = relu(LN(bufA @ W2 + b2))
    gemm_epilogue_kernel<100, 7, 100, false, true, true, false>
        <<<gGrid, gBlk, 0, stream>>>(bufA, W2, nullptr, nullptr, b2, g2, be2, bufB, rowTiles);

    // degree / symmetric normalization (shared by both GCN convs)
    fill_kernel<<<(N + TPB - 1) / TPB, TPB, 0, stream>>>(dinv, 1.0f, N);       // self-loop
    deg_kernel<<<(E + TPB - 1) / TPB, TPB, 0, stream>>>(tgt, dinv, E);
    rsqrt_kernel<<<(N + TPB - 1) / TPB, TPB, 0, stream>>>(dinv, N);

    // RGCN: bufC = segment_sum(bufB[src]);  bufA = relu(bufB@Wself + bufC@Wrel + rb)
    fill_kernel<<<(nElem + TPB - 1) / TPB, TPB, 0, stream>>>(bufC, 0.0f, nElem);
    scatter_add_kernel<<<((size_t)E * 32 + TPB - 1) / TPB, TPB, 0, stream>>>(bufB, src, tgt, bufC, E);
    gemm_epilogue_kernel<100, 7, 100, true, false, true, false>
        <<<gGrid, gBlk, 0, stream>>>(bufB, Wself, bufC, Wrel, rb, nullptr, nullptr, bufA, rowTiles);

    // GCNConv1: bufC = dinv^2*bufA + sum_e norm*bufA[src];  bufB = relu(bufC@c1W + c1b)
    scale_self_kernel<<<(nElem + TPB - 1) / TPB, TPB, 0, stream>>>(bufA, dinv, bufC, nElem);
    scatter_norm_kernel<<<((size_t)E * 32 + TPB - 1) / TPB, TPB, 0, stream>>>(bufA, src, tgt, dinv, bufC, E);
    gemm_epilogue_kernel<100, 7, 100, false, false, true, false>
        <<<gGrid, gBlk, 0, stream>>>(bufC, c1W, nullptr, nullptr, c1b, nullptr, nullptr, bufB, rowTiles);

    // GCNConv2 + log_softmax: out = log_softmax((dinv^2*bufB + agg)@c2W + c2b)
    scale_self_kernel<<<(nElem + TPB - 1) / TPB, TPB, 0, stream>>>(bufB, dinv, bufC, nElem);
    scatter_norm_kernel<<<((size_t)E * 32 + TPB - 1) / TPB, TPB, 0, stream>>>(bufB, src, tgt, dinv, bufC, E);
    gemm_epilogue_kernel<100, 4, 64, false, false, false, true>
        <<<gGrid, gBlk, 0, stream>>>(bufC, c2W, nullptr, nullptr, c2b, nullptr, nullptr,
                                     (float*)d_out, rowTiles);

    (void)n_in; (void)out_size; (void)ws_size; (void)DOUT;
}